// LeGNN_77223511982150
// MI455X (gfx1250) — compile-verified
//
#include <hip/hip_runtime.h>

typedef float v2f __attribute__((ext_vector_type(2)));
typedef float v8f __attribute__((ext_vector_type(8)));

#define DIN 128
#define H   64
#define NLAYERS 2
#define WS  72   // LDS stride for weight tiles: 2*72 mod 64 = 16 -> the two
                 // lane-halves (K and K+2 rows) hit disjoint bank sets

static __device__ __forceinline__ v8f wmma4(v2f a, v2f b, v8f c) {
  // D = A(16x4 f32) * B(4x16 f32) + C(16x16 f32)
  return __builtin_amdgcn_wmma_f32_16x16x4_f32(false, a, false, b, (short)0, c,
                                               false, false);
}

// stage a K x H row-major weight into LDS with padded stride WS
static __device__ __forceinline__ void stage_weight(float* dst, const float* W,
                                                    int K, int tid) {
  for (int i4 = tid; i4 < K * H / 4; i4 += 256) {
    const int r = (i4 * 4) / H;
    const int c = (i4 * 4) % H;
    const float4 v = *(const float4*)(W + i4 * 4);
    float* d = dst + r * WS + c;
    d[0] = v.x; d[1] = v.y; d[2] = v.z; d[3] = v.w;
  }
}

// ---------------------------------------------------------------------------
// h = x @ W_emb + b_emb          (N x 128) @ (128 x 64)
// one wave -> one 16x64 output tile; 16x128 A-tile staged in LDS (pad 129)
// ---------------------------------------------------------------------------
__global__ __launch_bounds__(256)
void k_node_emb(const float* __restrict__ x, const float* __restrict__ W,
                const float* __restrict__ bias, float* __restrict__ h, int n)
{
  __shared__ float tW[DIN * WS];                 // 36.9 KB
  __shared__ float lds[8][16 * (DIN + 1)];       // 66 KB
  const int tid  = threadIdx.x;
  const int lane = tid & 31;
  const int wave = tid >> 5;
  const int row0 = (blockIdx.x * 8 + wave) * 16;
  const bool full = (row0 + 16 <= n);
  float* tA = lds[wave];

  stage_weight(tW, W, DIN, tid);

  for (int i = lane; i < 16 * (DIN / 4); i += 32) {
    const int r  = i / (DIN / 4);
    const int c4 = (i % (DIN / 4)) * 4;
    float4 v = make_float4(0.f, 0.f, 0.f, 0.f);
    if (full || row0 + r < n)
      v = *(const float4*)(x + (size_t)(row0 + r) * DIN + c4);
    float* d = tA + r * (DIN + 1) + c4;
    d[0] = v.x; d[1] = v.y; d[2] = v.z; d[3] = v.w;
  }
  __syncthreads();

  const int m  = lane & 15;          // M row (A) / N col (B,C)
  const int kh = (lane >> 4) * 2;    // K sub-offset from lane half
  v8f acc[4];
#pragma unroll
  for (int t = 0; t < 4; ++t) {
    const float bv = bias[t * 16 + m];
#pragma unroll
    for (int j = 0; j < 8; ++j) acc[t][j] = bv;
  }
  for (int k = 0; k < DIN; k += 4) {
    v2f a;
    a.x = tA[m * (DIN + 1) + k + kh];
    a.y = tA[m * (DIN + 1) + k + kh + 1];
#pragma unroll
    for (int t = 0; t < 4; ++t) {
      v2f b;
      b.x = tW[(k + kh) * WS + t * 16 + m];
      b.y = tW[(k + kh + 1) * WS + t * 16 + m];
      acc[t] = wmma4(a, b, acc[t]);
    }
  }
  const int mbase = (lane >> 4) * 8;
  if (full) {
#pragma unroll
    for (int t = 0; t < 4; ++t)
#pragma unroll
      for (int j = 0; j < 8; ++j)
        h[(size_t)(row0 + mbase + j) * H + t * 16 + m] = acc[t][j];
  } else if (row0 < n) {
#pragma unroll
    for (int t = 0; t < 4; ++t)
#pragma unroll
      for (int j = 0; j < 8; ++j) {
        const int r = row0 + mbase + j;
        if (r < n) h[(size_t)r * H + t * 16 + m] = acc[t][j];
      }
  }
}

// ---------------------------------------------------------------------------
// a = h @ W1 + b1 ; b = h @ W2     (both from one staged A tile)
// ---------------------------------------------------------------------------
__global__ __launch_bounds__(256)
void k_layer_pre(const float* __restrict__ h, const float* __restrict__ W1,
                 const float* __restrict__ b1, const float* __restrict__ W2,
                 float* __restrict__ Aout, float* __restrict__ Bout, int n)
{
  __shared__ float tW1[H * WS];                  // 18.4 KB
  __shared__ float tW2[H * WS];                  // 18.4 KB
  __shared__ float lds[8][16 * (H + 1)];         // 33.3 KB
  const int tid  = threadIdx.x;
  const int lane = tid & 31;
  const int wave = tid >> 5;
  const int row0 = (blockIdx.x * 8 + wave) * 16;
  const bool full = (row0 + 16 <= n);
  float* tA = lds[wave];

  stage_weight(tW1, W1, H, tid);
  stage_weight(tW2, W2, H, tid);

  for (int i = lane; i < 16 * (H / 4); i += 32) {
    const int r  = i / (H / 4);
    const int c4 = (i % (H / 4)) * 4;
    float4 v = make_float4(0.f, 0.f, 0.f, 0.f);
    if (full || row0 + r < n)
      v = *(const float4*)(h + (size_t)(row0 + r) * H + c4);
    float* d = tA + r * (H + 1) + c4;
    d[0] = v.x; d[1] = v.y; d[2] = v.z; d[3] = v.w;
  }
  __syncthreads();

  const int m  = lane & 15;
  const int kh = (lane >> 4) * 2;
  v8f accA[4], accB[4];
#pragma unroll
  for (int t = 0; t < 4; ++t) {
    const float bv = b1[t * 16 + m];
#pragma unroll
    for (int j = 0; j < 8; ++j) { accA[t][j] = bv; accB[t][j] = 0.f; }
  }
  for (int k = 0; k < H; k += 4) {
    v2f a;
    a.x = tA[m * (H + 1) + k + kh];
    a.y = tA[m * (H + 1) + k + kh + 1];
#pragma unroll
    for (int t = 0; t < 4; ++t) {
      v2f b;
      b.x = tW1[(k + kh) * WS + t * 16 + m];
      b.y = tW1[(k + kh + 1) * WS + t * 16 + m];
      accA[t] = wmma4(a, b, accA[t]);
      v2f b2;
      b2.x = tW2[(k + kh) * WS + t * 16 + m];
      b2.y = tW2[(k + kh + 1) * WS + t * 16 + m];
      accB[t] = wmma4(a, b2, accB[t]);
    }
  }
  const int mbase = (lane >> 4) * 8;
  if (full) {
#pragma unroll
    for (int t = 0; t < 4; ++t)
#pragma unroll
      for (int j = 0; j < 8; ++j) {
        const size_t o = (size_t)(row0 + mbase + j) * H + t * 16 + m;
        Aout[o] = accA[t][j];
        Bout[o] = accB[t][j];
      }
  } else if (row0 < n) {
#pragma unroll
    for (int t = 0; t < 4; ++t)
#pragma unroll
      for (int j = 0; j < 8; ++j) {
        const int r = row0 + mbase + j;
        if (r < n) {
          Aout[(size_t)r * H + t * 16 + m] = accA[t][j];
          Bout[(size_t)r * H + t * 16 + m] = accB[t][j];
        }
      }
  }
}

// ---------------------------------------------------------------------------
// edge scatter: agg[dst] += (a[src] - b[dst]) * w   (16 threads/edge, float4)
// a/b/agg are L2-resident (77 MB < 192 MB L2); atomics are native f32 adds.
// ---------------------------------------------------------------------------
__global__ __launch_bounds__(256)
void k_edges(const long long* __restrict__ ei, const float* __restrict__ wat,
             const float* __restrict__ A, const float* __restrict__ B,
             float* __restrict__ agg, int e_total)
{
  const int t = blockIdx.x * 256 + threadIdx.x;
  const int e = t >> 4;
  const int c = (t & 15) * 4;
  if (e >= e_total) return;
  __builtin_prefetch(ei + e + 4096, 0, 1);           // global_prefetch_b8
  const long long s = ei[e];
  const long long d = ei[(size_t)e_total + e];
  const float w = wat[e];
  const float4 av = *(const float4*)(A + (size_t)s * H + c);
  const float4 bv = *(const float4*)(B + (size_t)d * H + c);
  float* p = agg + (size_t)d * H + c;
  unsafeAtomicAdd(p + 0, (av.x - bv.x) * w);
  unsafeAtomicAdd(p + 1, (av.y - bv.y) * w);
  unsafeAtomicAdd(p + 2, (av.z - bv.z) * w);
  unsafeAtomicAdd(p + 3, (av.w - bv.w) * w);
}

// ---------------------------------------------------------------------------
// h_out = relu(agg + h @ W3 + b3)   (in-place safe: wave reads only its rows)
// ---------------------------------------------------------------------------
__global__ __launch_bounds__(256)
void k_layer_post(const float* __restrict__ h, const float* __restrict__ agg,
                  const float* __restrict__ W3, const float* __restrict__ b3,
                  float* __restrict__ out, int n)
{
  __shared__ float tW[H * WS];
  __shared__ float lds[8][16 * (H + 1)];
  const int tid  = threadIdx.x;
  const int lane = tid & 31;
  const int wave = tid >> 5;
  const int row0 = (blockIdx.x * 8 + wave) * 16;
  const bool full = (row0 + 16 <= n);
  float* tA = lds[wave];

  stage_weight(tW, W3, H, tid);

  for (int i = lane; i < 16 * (H / 4); i += 32) {
    const int r  = i / (H / 4);
    const int c4 = (i % (H / 4)) * 4;
    float4 v = make_float4(0.f, 0.f, 0.f, 0.f);
    if (full || row0 + r < n)
      v = *(const float4*)(h + (size_t)(row0 + r) * H + c4);
    float* d = tA + r * (H + 1) + c4;
    d[0] = v.x; d[1] = v.y; d[2] = v.z; d[3] = v.w;
  }
  __syncthreads();

  const int m  = lane & 15;
  const int kh = (lane >> 4) * 2;
  v8f acc[4];
#pragma unroll
  for (int t = 0; t < 4; ++t) {
    const float bv = b3[t * 16 + m];
#pragma unroll
    for (int j = 0; j < 8; ++j) acc[t][j] = bv;
  }
  for (int k = 0; k < H; k += 4) {
    v2f a;
    a.x = tA[m * (H + 1) + k + kh];
    a.y = tA[m * (H + 1) + k + kh + 1];
#pragma unroll
    for (int t = 0; t < 4; ++t) {
      v2f b;
      b.x = tW[(k + kh) * WS + t * 16 + m];
      b.y = tW[(k + kh + 1) * WS + t * 16 + m];
      acc[t] = wmma4(a, b, acc[t]);
    }
  }
  const int mbase = (lane >> 4) * 8;
  if (full) {
#pragma unroll
    for (int t = 0; t < 4; ++t)
#pragma unroll
      for (int j = 0; j < 8; ++j) {
        const size_t o = (size_t)(row0 + mbase + j) * H + t * 16 + m;
        const float v = acc[t][j] + agg[o];
        out[o] = v > 0.f ? v : 0.f;
      }
  } else if (row0 < n) {
#pragma unroll
    for (int t = 0; t < 4; ++t)
#pragma unroll
      for (int j = 0; j < 8; ++j) {
        const int r = row0 + mbase + j;
        if (r < n) {
          const size_t o = (size_t)r * H + t * 16 + m;
          const float v = acc[t][j] + agg[o];
          out[o] = v > 0.f ? v : 0.f;
        }
      }
  }
}

// ---------------------------------------------------------------------------
extern "C" void kernel_launch(void* const* d_in, const int* in_sizes, int n_in,
                              void* d_out, int out_size, void* d_ws, size_t ws_size,
                              hipStream_t stream)
{
  const float*     x    = (const float*)d_in[0];
  const long long* ei   = (const long long*)d_in[1];
  const float*     ea   = (const float*)d_in[2];
  /* d_in[3] = batch (unused) */
  const float*     Wemb = (const float*)d_in[4];
  const float*     bemb = (const float*)d_in[5];
  const float*     W1   = (const float*)d_in[6];
  const float*     b1   = (const float*)d_in[7];
  const float*     W2   = (const float*)d_in[8];
  const float*     W3   = (const float*)d_in[9];
  const float*     b3   = (const float*)d_in[10];
  float* out = (float*)d_out;

  const int n = in_sizes[0] / DIN;   // N nodes
  const int e = in_sizes[2];         // E edges

  float* h    = (float*)d_ws;
  float* Abuf = h    + (size_t)n * H;
  float* Bbuf = Abuf + (size_t)n * H;
  float* agg  = Bbuf + (size_t)n * H;

  const int tiles = (n + 15) / 16;
  dim3 blk(256);
  dim3 grd((tiles + 7) / 8);
  const int eblk = (e * 16 + 255) / 256;

  k_node_emb<<<grd, blk, 0, stream>>>(x, Wemb, bemb, h, n);
  for (int l = 0; l < NLAYERS; ++l) {
    k_layer_pre<<<grd, blk, 0, stream>>>(h, W1 + l * H * H, b1 + l * H,
                                         W2 + l * H * H, Abuf, Bbuf, n);
    hipMemsetAsync(agg, 0, (size_t)n * H * sizeof(float), stream);
    k_edges<<<dim3(eblk), blk, 0, stream>>>(ei, ea, Abuf, Bbuf, agg, e);
    float* hout = (l == NLAYERS - 1) ? out : h;
    k_layer_post<<<grd, blk, 0, stream>>>(h, agg, W3 + l * H * H, b3 + l * H,
                                          hout, n);
  }
}